// GCNII_76278619177598
// MI455X (gfx1250) — compile-verified
//
#include <hip/hip_runtime.h>
#include <math.h>

typedef float v2f __attribute__((ext_vector_type(2)));
typedef float v8f __attribute__((ext_vector_type(8)));

#define IN_DIM   512
#define HID      64
#define NCLS     7
#define ALPHA_C  0.1f
#define XS_PITCH 516   // 516 % 64 == 4 -> conflict-free column reads of 32-row tile
#define OS_PITCH 68    // 68  % 64 == 4 -> same property

// ---------------- prologue kernels ----------------

__global__ __launch_bounds__(256) void k_deg_init(float* __restrict__ deg, int N) {
    int i = blockIdx.x * 256 + threadIdx.x;
    if (i < N) deg[i] = 1.0f;  // self-loop contribution
}

__global__ __launch_bounds__(256) void k_edge_prep(const long long* __restrict__ ei, int E,
                                                   int* __restrict__ src, int* __restrict__ dst,
                                                   float* __restrict__ deg) {
    int e = blockIdx.x * 256 + threadIdx.x;
    if (e >= E) return;
    int s = (int)ei[e];
    int d = (int)ei[(size_t)E + e];
    src[e] = s;
    dst[e] = d;
    unsafeAtomicAdd(&deg[d], 1.0f);   // hardware global_atomic_add_f32
}

__global__ __launch_bounds__(256) void k_rsqrt(float* __restrict__ deg, int N) {
    int i = blockIdx.x * 256 + threadIdx.x;
    if (i < N) deg[i] = rsqrtf(deg[i]);   // deg >= 1 always (self-loops)
}

__global__ __launch_bounds__(256) void k_norm(const int* __restrict__ src, const int* __restrict__ dst,
                                              const float* __restrict__ dinv, float* __restrict__ nrm,
                                              int E) {
    int e = blockIdx.x * 256 + threadIdx.x;
    if (e < E) nrm[e] = dinv[src[e]] * dinv[dst[e]];
}

// ---------------- input GEMM: h0 = relu(x @ W0 + b0), fp32 WMMA ----------------
// block = 256 threads = 8 waves; each block computes a 32x64 tile of h0.
// wave w: mTile = w/4 (0..1), nTile = w%4 (0..3); each wave owns a 16x16 C tile.

__global__ __launch_bounds__(256) void k_gemm_in(const float* __restrict__ x,
                                                 const float* __restrict__ W0,
                                                 const float* __restrict__ b0,
                                                 float* __restrict__ h0, int N) {
    __shared__ float xs[32 * XS_PITCH];
    const int tid = threadIdx.x;
    const int mBlock = blockIdx.x * 32;

    // stage the 32x512 fp32 x-tile into LDS with gfx1250 async global->LDS copies
    // (GLOBAL_LOAD_ASYNC_TO_LDS_B128, tracked by ASYNCcnt; no VGPR round-trip)
    for (int i = tid; i < 32 * (IN_DIM / 4); i += 256) {
        int r  = i >> 7;             // 128 float4 per row
        int c4 = (i & 127) * 4;
        int row = mBlock + r; if (row >= N) row = N - 1;
        unsigned long long gp = (unsigned long long)(uintptr_t)(x + (size_t)row * IN_DIM + c4);
        unsigned int lds = (unsigned int)(uintptr_t)(&xs[r * XS_PITCH + c4]); // addr[31:0] = LDS offset
        asm volatile("global_load_async_to_lds_b128 %0, %1, off"
                     :: "v"(lds), "v"(gp) : "memory");
    }
    asm volatile("s_wait_asynccnt 0x0" ::: "memory");
    __syncthreads();

    const int wave = tid >> 5, lane = tid & 31;
    const int mTile = wave >> 2, nTile = wave & 3;
    const int half = lane >> 4, l16 = lane & 15;

    // A 16x4 layout: lanes 0-15 hold K={k0,k0+1} of row M=l16; lanes 16-31 hold K={k0+2,k0+3}
    const float* arow = &xs[(mTile * 16 + l16) * XS_PITCH + half * 2];
    // B 4x16 layout (mirror over N): lane holds column N=l16, rows K={k0,k0+1} or {k0+2,k0+3}
    const float* bcol = W0 + (size_t)(half * 2) * HID + (nTile * 16 + l16);

    v8f c = {};
#pragma unroll 4
    for (int k0 = 0; k0 < IN_DIM; k0 += 4) {
        v2f a, b;
        a.x = arow[k0];
        a.y = arow[k0 + 1];
        b.x = bcol[(size_t)k0 * HID];
        b.y = bcol[(size_t)(k0 + 1) * HID];
        c = __builtin_amdgcn_wmma_f32_16x16x4_f32(false, a, false, b, (short)0, c, false, false);
    }

    // C layout: VGPR q -> lanes 0-15: (M=q, N=lane); lanes 16-31: (M=8+q, N=lane-16)
    const int cn = nTile * 16 + l16;
    const float bias = b0[cn];
    float* outp = h0 + (size_t)(mBlock + mTile * 16 + half * 8) * HID + cn;
    if (mBlock + 32 <= N) {           // full tile: branch-free store clause
#pragma unroll
        for (int q = 0; q < 8; ++q) {
            float r = c[q] + bias;
            outp[(size_t)q * HID] = r > 0.f ? r : 0.f;
        }
    } else {                          // ragged tail (not hit when N % 32 == 0)
#pragma unroll
        for (int q = 0; q < 8; ++q) {
            int row = mBlock + mTile * 16 + q + half * 8;
            if (row < N) {
                float r = c[q] + bias;
                h0[(size_t)row * HID + cn] = r > 0.f ? r : 0.f;
            }
        }
    }
}

// ---------------- SpMM: agg = A_hat @ h ----------------

__global__ __launch_bounds__(256) void k_selfloop(const float* __restrict__ hin,
                                                  const float* __restrict__ dinv,
                                                  float* __restrict__ agg, int N) {
    int gid = blockIdx.x * 256 + threadIdx.x;   // N*16 threads, one float4 each
    int node = gid >> 4;
    if (node >= N) return;
    int c4 = (gid & 15) * 4;
    float d = dinv[node];
    float s = d * d;                             // self-loop norm = dinv^2
    float4 hv = *(const float4*)(hin + (size_t)node * HID + c4);
    float4 r; r.x = s * hv.x; r.y = s * hv.y; r.z = s * hv.z; r.w = s * hv.w;
    *(float4*)(agg + (size_t)node * HID + c4) = r;
}

__global__ __launch_bounds__(256) void k_scatter(const int* __restrict__ src, const int* __restrict__ dst,
                                                 const float* __restrict__ nrm,
                                                 const float* __restrict__ hin,
                                                 float* __restrict__ agg, int E) {
    int gid = blockIdx.x * 256 + threadIdx.x;   // E*16 threads; 16 lanes cover a 64-float row
    int e = gid >> 4;
    if (e >= E) return;
    int c4 = (gid & 15) * 4;
    int s = src[e], d = dst[e];
    float w = nrm[e];
    float4 hv = *(const float4*)(hin + (size_t)s * HID + c4);
    float* p = agg + (size_t)d * HID + c4;
    unsafeAtomicAdd(p + 0, w * hv.x);
    unsafeAtomicAdd(p + 1, w * hv.y);
    unsafeAtomicAdd(p + 2, w * hv.z);
    unsafeAtomicAdd(p + 3, w * hv.w);
}

// ---------------- fused layer: h = relu(beta*(out@W) + (1-beta)*out), out = .9*agg + .1*h0 ----

__global__ __launch_bounds__(256) void k_layer(const float* __restrict__ agg,
                                               const float* __restrict__ h0,
                                               const float* __restrict__ W,
                                               float* __restrict__ hout,
                                               float beta, int N) {
    __shared__ float outs[32 * OS_PITCH];
    const int tid = threadIdx.x;
    const int mBlock = blockIdx.x * 32;

    // phase 1: out tile (32x64) into LDS  (blend needs VALU, so regular path)
    for (int i = tid; i < 32 * (HID / 4); i += 256) {
        int r  = i >> 4;
        int c4 = (i & 15) * 4;
        int row = mBlock + r; if (row >= N) row = N - 1;
        size_t g = (size_t)row * HID + c4;
        float4 a4 = *(const float4*)(agg + g);
        float4 h4 = *(const float4*)(h0 + g);
        float* p = &outs[r * OS_PITCH + c4];
        p[0] = (1.f - ALPHA_C) * a4.x + ALPHA_C * h4.x;
        p[1] = (1.f - ALPHA_C) * a4.y + ALPHA_C * h4.y;
        p[2] = (1.f - ALPHA_C) * a4.z + ALPHA_C * h4.z;
        p[3] = (1.f - ALPHA_C) * a4.w + ALPHA_C * h4.w;
    }
    __syncthreads();

    const int wave = tid >> 5, lane = tid & 31;
    const int mTile = wave >> 2, nTile = wave & 3;
    const int half = lane >> 4, l16 = lane & 15;

    const float* arow = &outs[(mTile * 16 + l16) * OS_PITCH + half * 2];
    const float* bcol = W + (size_t)(half * 2) * HID + (nTile * 16 + l16);

    v8f c = {};
#pragma unroll
    for (int k0 = 0; k0 < HID; k0 += 4) {
        v2f a, b;
        a.x = arow[k0];
        a.y = arow[k0 + 1];
        b.x = bcol[(size_t)k0 * HID];
        b.y = bcol[(size_t)(k0 + 1) * HID];
        c = __builtin_amdgcn_wmma_f32_16x16x4_f32(false, a, false, b, (short)0, c, false, false);
    }

    const int cn = nTile * 16 + l16;
    const float g1 = 1.f - beta;
    const int rbase = mTile * 16 + half * 8;
    float* outp = hout + (size_t)(mBlock + rbase) * HID + cn;
    if (mBlock + 32 <= N) {           // full tile: branch-free
#pragma unroll
        for (int q = 0; q < 8; ++q) {
            float o = outs[(rbase + q) * OS_PITCH + cn];
            float r = beta * c[q] + g1 * o;
            outp[(size_t)q * HID] = r > 0.f ? r : 0.f;
        }
    } else {
#pragma unroll
        for (int q = 0; q < 8; ++q) {
            int rl = rbase + q;
            int row = mBlock + rl;
            if (row < N) {
                float o = outs[rl * OS_PITCH + cn];
                float r = beta * c[q] + g1 * o;
                hout[(size_t)row * HID + cn] = r > 0.f ? r : 0.f;
            }
        }
    }
}

// ---------------- output GEMM: out = h @ Wout + bout (N x 7) ----------------

__global__ __launch_bounds__(256) void k_out(const float* __restrict__ h,
                                             const float* __restrict__ Wout,
                                             const float* __restrict__ bout,
                                             float* __restrict__ out, int N) {
    __shared__ float ws[HID * NCLS];
    __shared__ float bs[NCLS];
    for (int i = threadIdx.x; i < HID * NCLS; i += 256) ws[i] = Wout[i];
    if (threadIdx.x < NCLS) bs[threadIdx.x] = bout[threadIdx.x];
    __syncthreads();

    int n = blockIdx.x * 256 + threadIdx.x;
    if (n >= N) return;
    float acc[NCLS];
#pragma unroll
    for (int cc = 0; cc < NCLS; ++cc) acc[cc] = bs[cc];
    const float* hr = h + (size_t)n * HID;
    for (int k = 0; k < HID; ++k) {
        float hv = hr[k];
#pragma unroll
        for (int cc = 0; cc < NCLS; ++cc) acc[cc] += hv * ws[k * NCLS + cc];
    }
#pragma unroll
    for (int cc = 0; cc < NCLS; ++cc) out[(size_t)n * NCLS + cc] = acc[cc];
}

// ---------------- host-side orchestration ----------------

extern "C" void kernel_launch(void* const* d_in, const int* in_sizes, int n_in,
                              void* d_out, int out_size, void* d_ws, size_t ws_size,
                              hipStream_t stream) {
    const float*     x    = (const float*)d_in[0];
    const long long* ei   = (const long long*)d_in[1];   // int64 edge_index [2, E]
    const float*     W0   = (const float*)d_in[2];
    const float*     b0   = (const float*)d_in[3];
    const float*     Wl   = (const float*)d_in[4];       // [L, 64, 64]
    const float*     Wout = (const float*)d_in[5];
    const float*     bout = (const float*)d_in[6];

    const int N = in_sizes[0] / IN_DIM;
    const int E = in_sizes[1] / 2;
    const int L = in_sizes[4] / (HID * HID);

    // workspace carving (~116 MB total; hot set fits MI455X's 192 MB L2)
    char* ws = (char*)d_ws;
    size_t off = 0;
    auto carve = [&](size_t bytes) { char* p = ws + off; off += (bytes + 255) & ~(size_t)255; return p; };
    int*   src32 = (int*)  carve((size_t)E * 4);
    int*   dst32 = (int*)  carve((size_t)E * 4);
    float* nrm   = (float*)carve((size_t)E * 4);
    float* dinv  = (float*)carve((size_t)N * 4);        // deg, then rsqrt in place
    float* h0    = (float*)carve((size_t)N * HID * 4);
    float* h     = (float*)carve((size_t)N * HID * 4);
    float* agg   = (float*)carve((size_t)N * HID * 4);
    (void)ws_size; (void)n_in; (void)out_size;

    const int nB256 = (N + 255) / 256;
    const int eB256 = (E + 255) / 256;

    // degree / normalization prologue
    k_deg_init<<<nB256, 256, 0, stream>>>(dinv, N);
    k_edge_prep<<<eB256, 256, 0, stream>>>(ei, E, src32, dst32, dinv);
    k_rsqrt<<<nB256, 256, 0, stream>>>(dinv, N);
    k_norm<<<eB256, 256, 0, stream>>>(src32, dst32, dinv, nrm, E);

    // h0 = relu(x @ W0 + b0)
    k_gemm_in<<<(N + 31) / 32, 256, 0, stream>>>(x, W0, b0, h0, N);

    // 32 GCNII layers
    const int slB = (int)(((size_t)N * 16 + 255) / 256);
    const int scB = (int)(((size_t)E * 16 + 255) / 256);
    for (int l = 0; l < L; ++l) {
        float beta = logf(0.5f / (float)(l + 1) + 1.0f);
        const float* hin = (l == 0) ? h0 : h;
        k_selfloop<<<slB, 256, 0, stream>>>(hin, dinv, agg, N);
        k_scatter<<<scB, 256, 0, stream>>>(src32, dst32, nrm, hin, agg, E);
        k_layer<<<(N + 31) / 32, 256, 0, stream>>>(agg, h0, Wl + (size_t)l * HID * HID, h, beta, N);
    }

    // final projection
    k_out<<<nB256, 256, 0, stream>>>(h, Wout, bout, (float*)d_out, N);
}